// KernelBlock_7387343749286
// MI455X (gfx1250) — compile-verified
//
#include <hip/hip_runtime.h>

// K[b,t,s] = F[b,t,:]·F[b,s,:] + const
//          + exp(-(|F_t|^2 + |F_s|^2 - 2 F_t·F_s) * 0.5 / scale[s]^2)
//          + 1e-5 * (t==s)
// B=8, T=2048, C=128, fp32 in/out. Fused WMMA-f32 GEMM + epilogue.

typedef __attribute__((ext_vector_type(2))) float v2f;
typedef __attribute__((ext_vector_type(8))) float v8f;

#define TDIM 2048
#define CDIM 128
#define BT   128      // output tile per workgroup (BT x BT)
#define LDSTR 132     // padded LDS row stride in floats (132 % 64 == 4 -> conflict-free b64)
#define EPS_DIAG 1e-5f

__global__ __launch_bounds__(256)
void gram_rbf_kernel(const float* __restrict__ features,
                     const float* __restrict__ constp,
                     const float* __restrict__ scale,
                     float* __restrict__ out)
{
    __shared__ float lsA[BT * LDSTR];
    __shared__ float lsB[BT * LDSTR];
    __shared__ float sqA[BT];
    __shared__ float sqB[BT];

    const int b     = blockIdx.z;
    const int ibase = blockIdx.y * BT;   // output row block
    const int jbase = blockIdx.x * BT;   // output col block
    const int tid   = threadIdx.x;

    const float* fb = features + (size_t)b * TDIM * CDIM;

    // ---- stage two 128x128 fp32 row-blocks of F into LDS (float4 vectorized) ----
    {
        const int c4   = (tid & 31) * 4;   // float column, 32 float4 per 128-wide row
        const int rsub = tid >> 5;         // 0..7 rows per pass
        for (int rr = 0; rr < BT; rr += 8) {
            const int row = rr + rsub;
            float4 va = *(const float4*)(fb + (size_t)(ibase + row) * CDIM + c4);
            float4 vb = *(const float4*)(fb + (size_t)(jbase + row) * CDIM + c4);
            *(float4*)&lsA[row * LDSTR + c4] = va;
            *(float4*)&lsB[row * LDSTR + c4] = vb;
        }
    }
    __syncthreads();

    // ---- per-row sum of squares (RBF distance term), one row per thread ----
    {
        const int r = tid & (BT - 1);
        const float* rowp = (tid < BT) ? &lsA[r * LDSTR] : &lsB[r * LDSTR];
        float s = 0.f;
        #pragma unroll 8
        for (int c = 0; c < CDIM; ++c) { const float x = rowp[c]; s = fmaf(x, x, s); }
        if (tid < BT) sqA[r] = s; else sqB[r] = s;
    }
    __syncthreads();

    // ---- WMMA gram: wave owns 32x64 region = 2x4 tiles of 16x16 ----
    const int wave  = tid >> 5;
    const int lane  = tid & 31;
    const int lmod  = lane & 15;
    const int lhalf = lane >> 4;
    const int rbase = (wave & 3) * 32;    // row offset inside tile (4 row groups)
    const int cbase = (wave >> 2) * 64;   // col offset inside tile (2 col groups)

    v8f acc[2][4];
    #pragma unroll
    for (int m = 0; m < 2; ++m)
        #pragma unroll
        for (int n = 0; n < 4; ++n) { v8f z = {}; acc[m][n] = z; }

    // A 16x4 frag: lane holds A[M = lane%16][K = k0 + 2*lhalf + {0,1}]  (v2f)
    // B 4x16 frag: lane holds B[K = k0 + 2*lhalf + {0,1}][N = lane%16]
    //            = F_B[col-row = cbase + n*16 + lane%16][same K]  -> identical addressing
    const float* aptr = &lsA[(rbase + lmod) * LDSTR + 2 * lhalf];
    const float* bptr = &lsB[(cbase + lmod) * LDSTR + 2 * lhalf];

    #pragma unroll 4
    for (int k0 = 0; k0 < CDIM; k0 += 4) {
        v2f a0 = *(const v2f*)(aptr + k0);
        v2f a1 = *(const v2f*)(aptr + 16 * LDSTR + k0);
        v2f b0 = *(const v2f*)(bptr + k0);
        v2f b1 = *(const v2f*)(bptr + 16 * LDSTR + k0);
        v2f b2 = *(const v2f*)(bptr + 32 * LDSTR + k0);
        v2f b3 = *(const v2f*)(bptr + 48 * LDSTR + k0);

        acc[0][0] = __builtin_amdgcn_wmma_f32_16x16x4_f32(false, a0, false, b0, (short)0, acc[0][0], false, false);
        acc[0][1] = __builtin_amdgcn_wmma_f32_16x16x4_f32(false, a0, false, b1, (short)0, acc[0][1], false, false);
        acc[0][2] = __builtin_amdgcn_wmma_f32_16x16x4_f32(false, a0, false, b2, (short)0, acc[0][2], false, false);
        acc[0][3] = __builtin_amdgcn_wmma_f32_16x16x4_f32(false, a0, false, b3, (short)0, acc[0][3], false, false);
        acc[1][0] = __builtin_amdgcn_wmma_f32_16x16x4_f32(false, a1, false, b0, (short)0, acc[1][0], false, false);
        acc[1][1] = __builtin_amdgcn_wmma_f32_16x16x4_f32(false, a1, false, b1, (short)0, acc[1][1], false, false);
        acc[1][2] = __builtin_amdgcn_wmma_f32_16x16x4_f32(false, a1, false, b2, (short)0, acc[1][2], false, false);
        acc[1][3] = __builtin_amdgcn_wmma_f32_16x16x4_f32(false, a1, false, b3, (short)0, acc[1][3], false, false);
    }

    // ---- fused epilogue: linear + RBF + eps*I, then store ----
    // C/D layout: VGPR v -> M = v + 8*lhalf, N = lmod
    const float constv = constp[0];
    float* outb = out + (size_t)b * TDIM * TDIM;

    #pragma unroll
    for (int n = 0; n < 4; ++n) {
        const int scol   = cbase + n * 16 + lmod;  // col inside tile
        const int s      = jbase + scol;           // global col
        const float sc   = scale[s];
        const float c2   = 0.5f / (sc * sc);
        const float sqs  = sqB[scol];
        #pragma unroll
        for (int m = 0; m < 2; ++m) {
            #pragma unroll
            for (int v = 0; v < 8; ++v) {
                const int mrow = rbase + m * 16 + v + 8 * lhalf; // row inside tile
                const int t    = ibase + mrow;                    // global row
                const float g  = acc[m][n][v];
                const float d  = sqA[mrow] + sqs - 2.0f * g;
                float kv = g + constv + expf(-d * c2);
                if (t == s) kv += EPS_DIAG;
                outb[(size_t)t * TDIM + s] = kv;
            }
        }
    }
}

extern "C" void kernel_launch(void* const* d_in, const int* in_sizes, int n_in,
                              void* d_out, int out_size, void* d_ws, size_t ws_size,
                              hipStream_t stream) {
    const float* features = (const float*)d_in[0];  // [8, 2048, 128] f32
    const float* constp   = (const float*)d_in[1];  // [1] f32
    const float* scale    = (const float*)d_in[2];  // [2048] f32
    float* out            = (float*)d_out;          // [8, 2048, 2048] f32

    dim3 grid(TDIM / BT, TDIM / BT, 8);
    dim3 block(256);
    gram_rbf_kernel<<<grid, block, 0, stream>>>(features, constp, scale, out);
}